// CrossFunctionsLoss_78125455114760
// MI455X (gfx1250) — compile-verified
//
#include <hip/hip_runtime.h>
#include <math.h>

typedef __attribute__((ext_vector_type(2))) float v2f;
typedef __attribute__((ext_vector_type(8))) float v8f;

#define NN 4096
#define DD 128

// workspace layout (floats)
#define N_GEMM_PART 6144      // ws[0 .. 6143]      : per-block GEMM-loss partials
#define WS_BQC      6144      // ws[6144 .. 6399]   : 128 FP-B sq partials, 128 FM-B sq partials
#define WS_FDC      6400      // ws[6400 .. 6655]   : 256 squared row sums (128 FP + 128 FM)

// ---------------------------------------------------------------------------
// Kernel 1: the three N x N Gram matrices via fp32 WMMA + streamed loss
// Each wave computes a 32x32 output block (2x2 grid of 16x16 WMMA tiles).
// 49152 waves total = 3 GEMMs * (128*128) 32x32-blocks; 8 waves / block.
// ---------------------------------------------------------------------------
__global__ __launch_bounds__(256) void gemm_loss_kernel(
    const float* __restrict__ SU, const float* __restrict__ SP,
    const float* __restrict__ SM, const float* __restrict__ FP,
    const float* __restrict__ FM, float* __restrict__ partials)
{
    __shared__ float red[8];

    const int lane = threadIdx.x & 31;
    const int wave = threadIdx.x >> 5;
    const unsigned w = blockIdx.x * 8u + wave;      // 0 .. 49151
    const unsigned g = w >> 14;                     // which GEMM: 0=UP*M, 1=P*P, 2=M*M
    const unsigned t = w & 16383u;
    const int i0 = (int)(t >> 7) * 32;              // row block
    const int j0 = (int)(t & 127) * 32;             // col block

    const float* Amat = (g == 2) ? FM : FP;
    const float* Bmat = (g == 1) ? FP : FM;
    const float* Smat = (g == 0) ? SU : ((g == 1) ? SP : SM);
    const float eps = (g == 0) ? 0.0f : 1e-8f;

    // WMMA f32 16x16x4 fragment addressing (wave32):
    //  A (16x4): lane m=lane&15 holds row M=m; K = 2*(lane>>4) + vgpr
    //  B (4x16): lane holds col N=m;           K = 2*(lane>>4) + vgpr
    const int m  = lane & 15;
    const int hi = lane >> 4;

    v8f acc[2][2] = {};   // 2x2 tiles of 16x16 f32 accumulators (32 VGPRs)

    const float* ap = Amat + (size_t)(2 * hi) * NN + i0 + m;
    const float* bp = Bmat + (size_t)(2 * hi) * NN + j0 + m;

    #pragma unroll 4
    for (int d0 = 0; d0 < DD; d0 += 4) {
        v2f a0, a1, b0, b1;
        a0.x = ap[0];        a0.y = ap[NN];
        a1.x = ap[16];       a1.y = ap[NN + 16];
        b0.x = bp[0];        b0.y = bp[NN];
        b1.x = bp[16];       b1.y = bp[NN + 16];
        ap += 4 * NN;
        bp += 4 * NN;
        acc[0][0] = __builtin_amdgcn_wmma_f32_16x16x4_f32(false, a0, false, b0,
                        (short)0, acc[0][0], false, false);
        acc[0][1] = __builtin_amdgcn_wmma_f32_16x16x4_f32(false, a0, false, b1,
                        (short)0, acc[0][1], false, false);
        acc[1][0] = __builtin_amdgcn_wmma_f32_16x16x4_f32(false, a1, false, b0,
                        (short)0, acc[1][0], false, false);
        acc[1][1] = __builtin_amdgcn_wmma_f32_16x16x4_f32(false, a1, false, b1,
                        (short)0, acc[1][1], false, false);
    }

    // Loss over the 32x32 block. C/D layout: VGPR r holds row M = r + 8*hi, col = m.
    // S is streamed once (192 MB total) -> non-temporal loads to keep F in L2.
    float sum = 0.0f;
    #pragma unroll
    for (int ci = 0; ci < 2; ++ci) {
        #pragma unroll
        for (int cj = 0; cj < 2; ++cj) {
            const float* sp = Smat + (size_t)(i0 + 16 * ci + 8 * hi) * NN
                                   + (j0 + 16 * cj + m);
            #pragma unroll
            for (int r = 0; r < 8; ++r) {
                float s = __builtin_nontemporal_load(sp + (size_t)r * NN);
                float omega = acc[ci][cj][r] * 0.5f;
                sum += fmaf(-s, omega, logf(1.0f + omega + eps));
            }
        }
    }

    // wave32 reduction
    #pragma unroll
    for (int off = 16; off >= 1; off >>= 1)
        sum += __shfl_xor(sum, off, 32);
    if (lane == 0) red[wave] = sum;
    __syncthreads();
    if (threadIdx.x == 0) {
        float b = 0.0f;
        #pragma unroll
        for (int v = 0; v < 8; ++v) b += red[v];
        partials[blockIdx.x] = b;
    }
}

// ---------------------------------------------------------------------------
// Kernel 2: BQC partial sums of squares and FDC squared row sums.
// blocks 0..255   : (F - B)^2 partials  (0..127 -> FP, 128..255 -> FM)
// blocks 256..511 : squared row sums    (256..383 -> FP rows, 384..511 -> FM)
// ---------------------------------------------------------------------------
__global__ __launch_bounds__(256) void small_terms_kernel(
    const float* __restrict__ FP, const float* __restrict__ FM,
    const float* __restrict__ Bm, float* __restrict__ ws)
{
    __shared__ float red[256];
    const int b = blockIdx.x;
    const int t = threadIdx.x;
    float sum = 0.0f;

    if (b < 256) {
        const float* F = (b < 128) ? FP : FM;
        const int base = (b & 127) * 4096;          // 128 chunks cover D*N = 524288
        #pragma unroll
        for (int k = 0; k < 16; ++k) {
            int idx = base + t + k * 256;
            float d = F[idx] - Bm[idx];
            sum += d * d;
        }
    } else {
        const int r = b - 256;
        const float* F = (r < 128) ? FP : FM;
        const float* p = F + (size_t)(r & 127) * NN;
        #pragma unroll
        for (int k = 0; k < 16; ++k)
            sum += p[t + k * 256];
    }

    red[t] = sum;
    __syncthreads();
    for (int s = 128; s > 0; s >>= 1) {
        if (t < s) red[t] += red[t + s];
        __syncthreads();
    }
    if (t == 0) {
        if (b < 256) ws[WS_BQC + b] = red[0];
        else         ws[WS_FDC + (b - 256)] = red[0] * red[0];
    }
}

// ---------------------------------------------------------------------------
// Kernel 3: deterministic fixed-order final reduction + sqrt terms.
// ---------------------------------------------------------------------------
__global__ __launch_bounds__(256) void finalize_kernel(
    const float* __restrict__ ws, float* __restrict__ out)
{
    __shared__ float red[256];
    const int t = threadIdx.x;

    // GEMM-loss partials (6144)
    float sum = 0.0f;
    for (int i = t; i < N_GEMM_PART; i += 256) sum += ws[i];
    red[t] = sum;
    __syncthreads();
    for (int s = 128; s > 0; s >>= 1) { if (t < s) red[t] += red[t + s]; __syncthreads(); }
    float sumGemm = red[0];
    __syncthreads();

    // ||FP - B||^2 partials (128)
    red[t] = (t < 128) ? ws[WS_BQC + t] : 0.0f;
    __syncthreads();
    for (int s = 128; s > 0; s >>= 1) { if (t < s) red[t] += red[t + s]; __syncthreads(); }
    float sqP = red[0];
    __syncthreads();

    // ||FM - B||^2 partials (128)
    red[t] = (t < 128) ? ws[WS_BQC + 128 + t] : 0.0f;
    __syncthreads();
    for (int s = 128; s > 0; s >>= 1) { if (t < s) red[t] += red[t + s]; __syncthreads(); }
    float sqM = red[0];
    __syncthreads();

    // FDC squared row sums (256)
    red[t] = ws[WS_FDC + t];
    __syncthreads();
    for (int s = 128; s > 0; s >>= 1) { if (t < s) red[t] += red[t + s]; __syncthreads(); }
    float fdc = red[0];

    if (t == 0)
        out[0] = sumGemm + sqrtf(sqP) + sqrtf(sqM) + fdc;
}

// ---------------------------------------------------------------------------
extern "C" void kernel_launch(void* const* d_in, const int* in_sizes, int n_in,
                              void* d_out, int out_size, void* d_ws, size_t ws_size,
                              hipStream_t stream) {
    (void)in_sizes; (void)n_in; (void)out_size; (void)ws_size;
    const float* SU = (const float*)d_in[0];
    const float* SP = (const float*)d_in[1];
    const float* SM = (const float*)d_in[2];
    const float* FP = (const float*)d_in[3];
    const float* FM = (const float*)d_in[4];
    const float* Bm = (const float*)d_in[5];
    float* ws  = (float*)d_ws;
    float* out = (float*)d_out;

    gemm_loss_kernel<<<N_GEMM_PART, 256, 0, stream>>>(SU, SP, SM, FP, FM, ws);
    small_terms_kernel<<<512, 256, 0, stream>>>(FP, FM, Bm, ws);
    finalize_kernel<<<1, 256, 0, stream>>>(ws, out);
}